// DigiCaps_74268574483122
// MI455X (gfx1250) — compile-verified
//
#include <hip/hip_runtime.h>

typedef float v2f __attribute__((ext_vector_type(2)));
typedef float v4f __attribute__((ext_vector_type(4)));
typedef float v8f __attribute__((ext_vector_type(8)));

#define B_TOT  512
#define INC    1152
#define IND    8
#define NC     10
#define DC     16
#define ISPLIT 16                 // i-dimension split across blocks
#define ICHUNK (INC / ISPLIT)     // 72 input capsules per block
#define BT     16                 // batch tile per block (WMMA N dimension)
#define BGRP   (B_TOT / BT)       // 32 batch groups
#define NWAVE  10
#define NTHR   (NWAVE * 32)       // 320 threads: 10 waves -> <=3 waves/SIMD
#define GSZ    (NC * DC * BT)     // 2560 floats per block tile

__device__ __forceinline__ v8f wmma_f32_k4(v2f a, v2f b, v8f c) {
  // D = A(16x4 f32) * B(4x16 f32) + C(16x16 f32)
  return __builtin_amdgcn_wmma_f32_16x16x4_f32(
      false, a, false, b, (short)0, c, false, false);
}

// u_hat tile [16d x 16b]: U[d,b] = sum_k W[j,i,d,k] * x[b0+b, i, k]
__device__ __forceinline__ v8f compute_U(const float* __restrict__ W, int j, int i,
                                         int bcol, int klo, v2f bt0, v2f bt1) {
  const float* wp = W + (((size_t)j * INC + i) * DC + bcol) * IND + klo;
  v2f a0 = *(const v2f*)(wp);
  v2f a1 = *(const v2f*)(wp + 4);
  v8f u = {};
  u = wmma_f32_k4(a0, bt0, u);   // K = 0..3
  u = wmma_f32_k4(a1, bt1, u);   // K = 4..7
  return u;
}

// grid: (ISPLIT, BGRP) = 512 blocks; block: 320 threads = 10 waves (wave32)
__global__ __launch_bounds__(NTHR, 1)
void digicaps_pass_kernel(const float* __restrict__ X,        // [512,1152,8]
                          const float* __restrict__ W,        // [10,1152,16,8]
                          float* __restrict__ s_part_glob,    // [ISPLIT,512,10,16]
                          const float* __restrict__ vacc_glob,// [512,10,16]
                          int r) {
  __shared__ __align__(16) float s_lds[GSZ];     // s partial [j][d][b]
  __shared__ __align__(16) float vacc_lds[GSZ];  // Vacc [j][b][d]

  const int tid  = threadIdx.x;
  const int lane = tid & 31;
  const int lwv  = tid >> 5;           // wave id 0..9
  const int bcol = lane & 15;          // batch column (WMMA N index)
  const int hi   = lane >> 4;          // lane half
  const int dhi  = hi * 8;             // this half's 8 C-rows: d = dhi..dhi+7
  const int klo  = hi * 2;             // A/B operand K-pair base
  const int b0   = blockIdx.y * BT;
  const int i0   = blockIdx.x * ICHUNK;

  for (int t = tid; t < GSZ; t += NTHR) s_lds[t] = 0.f;
  if (r != 0) {
    // stage Vacc in LDS as [j][b][d] (d contiguous -> float4 lane loads)
    for (int t = tid; t < GSZ; t += NTHR) {
      int j = t >> 8, b = (t >> 4) & 15, d = t & 15;
      vacc_lds[t] = vacc_glob[((size_t)(b0 + b) * NC + j) * DC + d];
    }
  }
  __syncthreads();

  const float* xlane = X + (size_t)(b0 + bcol) * INC * IND + klo;

  v8f s_part[NC];
#pragma unroll
  for (int j = 0; j < NC; ++j) { v8f z = {}; s_part[j] = z; }

  if (r == 0) {
    // logits are all zero -> coupling coefficients are exactly 1/10
    const float c0 = 1.f / 10.f;
    for (int i = i0 + lwv; i < i0 + ICHUNK; i += NWAVE) {  // wave-uniform
      const float* xp = xlane + (size_t)i * IND;
      v2f bt0 = *(const v2f*)(xp);
      v2f bt1 = *(const v2f*)(xp + 4);
#pragma unroll
      for (int j = 0; j < NC; ++j) {
        v8f u = compute_U(W, j, i, bcol, klo, bt0, bt1);
#pragma unroll
        for (int q = 0; q < 8; ++q) s_part[j][q] += c0 * u[q];
      }
    }
  } else {
    for (int i = i0 + lwv; i < i0 + ICHUNK; i += NWAVE) {  // wave-uniform
      const float* xp = xlane + (size_t)i * IND;
      v2f bt0 = *(const v2f*)(xp);
      v2f bt1 = *(const v2f*)(xp + 4);

      // single pass over j: U[j] stays live; Vacc fetched from LDS
      v8f   U[NC];
      float bias[NC];
#pragma unroll
      for (int j = 0; j < NC; ++j) {
        U[j] = compute_U(W, j, i, bcol, klo, bt0, bt1);
        const float* vp = &vacc_lds[(j * 16 + bcol) * DC + dhi];
        v4f vlo = *(const v4f*)(vp);
        v4f vhi = *(const v4f*)(vp + 4);
        // adjacent-element sub-loops so v_pk_fma forms without cross-movs
        float p = 0.f;
#pragma unroll
        for (int q = 0; q < 4; ++q) p += vlo[q] * U[j][q];
#pragma unroll
        for (int q = 0; q < 4; ++q) p += vhi[q] * U[j][q + 4];
        p += __shfl_xor(p, 16, 32);       // combine d-halves
        bias[j] = p;
      }

      // per-lane softmax over 10 output capsules
      float m = bias[0];
#pragma unroll
      for (int j = 1; j < NC; ++j) m = fmaxf(m, bias[j]);
      float e[NC], esum = 0.f;
#pragma unroll
      for (int j = 0; j < NC; ++j) { e[j] = __expf(bias[j] - m); esum += e[j]; }
      float inv = 1.f / esum;

#pragma unroll
      for (int j = 0; j < NC; ++j) {
        float c = e[j] * inv;
#pragma unroll
        for (int q = 0; q < 8; ++q) s_part[j][q] += c * U[j][q];
      }
    }
  }

  // block-local reduction in LDS (ds_add_f32)
#pragma unroll
  for (int j = 0; j < NC; ++j)
#pragma unroll
    for (int q = 0; q < 8; ++q)
      atomicAdd(&s_lds[(j * DC + dhi + q) * BT + bcol], s_part[j][q]);
  __syncthreads();

  // store this block's partial s to its private i-split slot (no atomics)
  for (int t = tid; t < GSZ; t += NTHR) {
    int j = t >> 8, d = (t >> 4) & 15, bb = t & 15;   // t = (j*16 + d)*16 + bb
    s_part_glob[(((size_t)blockIdx.x * B_TOT + (b0 + bb)) * NC + j) * DC + d] =
        s_lds[t];
  }
}

// 5120 threads: one per (b, j). Sums i-split partials deterministically,
// squashes, updates Vacc (overwrites at r=0 -> no memset needed), emits v at r=2.
__global__ void digicaps_squash_kernel(const float* __restrict__ s_part_glob,
                                       float* __restrict__ vacc_glob,
                                       float* __restrict__ out, int r) {
  int idx = blockIdx.x * blockDim.x + threadIdx.x;
  if (idx >= B_TOT * NC) return;
  int b = idx / NC, j = idx % NC;

  float sv[DC];
#pragma unroll
  for (int d = 0; d < DC; ++d) sv[d] = 0.f;
  for (int is = 0; is < ISPLIT; ++is) {
    const float* sp = s_part_glob + (((size_t)is * B_TOT + b) * NC + j) * DC;
#pragma unroll
    for (int d = 0; d < DC; ++d) sv[d] += sp[d];
  }
  float sq = 0.f;
#pragma unroll
  for (int d = 0; d < DC; ++d) sq += sv[d] * sv[d];
  float scale = (sq / (1.f + sq)) / sqrtf(sq + 1e-7f);

  if (r < 2) {
    float* vp = vacc_glob + ((size_t)b * NC + j) * DC;
#pragma unroll
    for (int d = 0; d < DC; ++d)
      vp[d] = (r == 0) ? scale * sv[d] : vp[d] + scale * sv[d];
  } else {
#pragma unroll
    for (int d = 0; d < DC; ++d)
      out[((size_t)b * NC + j) * DC + d] = scale * sv[d];
  }
}

extern "C" void kernel_launch(void* const* d_in, const int* in_sizes, int n_in,
                              void* d_out, int out_size, void* d_ws, size_t ws_size,
                              hipStream_t stream) {
  const float* X = (const float*)d_in[0];   // inputs [512,1152,8] fp32
  const float* W = (const float*)d_in[1];   // W [10,1152,16,8] fp32
  float* out = (float*)d_out;               // v [512,10,16] fp32

  float* s_part_glob = (float*)d_ws;                                    // ISPLIT*512*160 floats
  float* vacc_glob   = s_part_glob + (size_t)ISPLIT * B_TOT * NC * DC;  // 512*160 floats
  // workspace use: (16+1)*512*160*4 B ~= 5.6 MB

  dim3 pgrid(ISPLIT, BGRP);                 // 16 x 32 = 512 blocks
  dim3 pblock(NTHR);                        // 320 threads = 10 waves
  dim3 qgrid((B_TOT * NC + 255) / 256);     // 20 blocks
  dim3 qblock(256);

  for (int r = 0; r < 3; ++r) {
    digicaps_pass_kernel<<<pgrid, pblock, 0, stream>>>(X, W, s_part_glob, vacc_glob, r);
    digicaps_squash_kernel<<<qgrid, qblock, 0, stream>>>(s_part_glob, vacc_glob, out, r);
  }
}